// GraphSAGE_Segmenter_35631048688034
// MI455X (gfx1250) — compile-verified
//
#include <hip/hip_runtime.h>

// GraphSAGE 3-layer forward for MI455X (gfx1250, wave32).
// Strategy:
//  - degree computed once (same edge list for all layers)
//  - segment-sum via float atomics, float4-granular edge parallelism
//  - fused (agg/deg)@Wl + b + h@Wr -> LayerNorm -> ReLU per layer using
//    V_WMMA_F32_16X16X4_F32 (full fp32 matrix path)
//  - final 64->2 layer scalar (too skinny for 16x16 tiles)

typedef __attribute__((ext_vector_type(2))) float v2f;
typedef __attribute__((ext_vector_type(8))) float v8f;

__global__ void zero_f32(float* __restrict__ p, long n) {
  long i = (long)blockIdx.x * blockDim.x + threadIdx.x;
  if (i < n) p[i] = 0.0f;
}

__global__ void deg_count(const long long* __restrict__ tgt, float* __restrict__ deg, int E) {
  int e = blockIdx.x * blockDim.x + threadIdx.x;
  if (e < E) atomicAdd(&deg[tgt[e]], 1.0f);
}

// One thread per (edge, 4-channel group): coalesced float4 gather of the
// source row, 4 global float atomics into the target row.
template <int C>
__global__ void scatter_add(const float* __restrict__ x, const long long* __restrict__ src,
                            const long long* __restrict__ tgt, float* __restrict__ agg, int E) {
  constexpr int G = C / 4;
  long t = (long)blockIdx.x * blockDim.x + threadIdx.x;
  long e = t / G;
  int g = (int)(t % G);
  if (e >= E) return;
  long s = src[e], d = tgt[e];
  const float4 v = *(const float4*)(x + s * (long)C + g * 4);
  float* base = agg + d * (long)C + g * 4;
  atomicAdd(base + 0, v.x);
  atomicAdd(base + 1, v.y);
  atomicAdd(base + 2, v.z);
  atomicAdd(base + 3, v.w);
}

// Fused SAGEConv combine + LayerNorm + ReLU.
// Block = 128 threads = 4 waves; each wave owns a 16-node block and produces
// all 64 output channels via 4 WMMA column tiles.
// WMMA f32 16x16x4 fragment layout (ISA 7.12.2):
//   A (16x4): lane&15 = M row; half-wave selects K pair {2h, 2h+1}
//   B (4x16): lane&15 = N col; half-wave selects K pair
//   C/D     : vgpr v, lane -> D[v + 8*half][lane&15]
template <int K>
__global__ __launch_bounds__(128) void sage_combine(
    const float* __restrict__ agg, const float* __restrict__ deg,
    const float* __restrict__ h, const float* __restrict__ Wl,
    const float* __restrict__ Wr, const float* __restrict__ bias,
    const float* __restrict__ gamma, const float* __restrict__ beta,
    float* __restrict__ out, int n) {
  constexpr int NW = 4;       // waves per block
  constexpr int STR = 68;     // LDS row stride (64 cols + pad, avoids bank conflicts)
  __shared__ float sWl[K * 64];
  __shared__ float sWr[K * 64];
  __shared__ float sOut[NW][16 * STR];
  __shared__ float sStat[NW][32];  // [0..15]=mu, [16..31]=rstd

  const int tid = threadIdx.x;
  for (int i = tid; i < K * 64; i += 128) {
    sWl[i] = Wl[i];
    sWr[i] = Wr[i];
  }
  __syncthreads();

  const int wave = tid >> 5;
  const int lane = tid & 31;
  const int hf = lane >> 4;   // half-wave (K-pair select)
  const int r = lane & 15;    // A: M row / B: N col
  const long node0 = ((long)blockIdx.x * NW + wave) * 16;
  const bool active = node0 < (long)n;  // N % 16 == 0 -> whole wave in/out
  const long nodeR = active ? (node0 + r) : (long)r;
  const float rdeg = 1.0f / fmaxf(deg[nodeR], 1.0f);
  const float* aRow = agg + nodeR * K;
  const float* hRow = h + nodeR * K;

  for (int t = 0; t < 4; ++t) {
    v8f c = {0.f, 0.f, 0.f, 0.f, 0.f, 0.f, 0.f, 0.f};
    const int cb = t * 16 + r;  // output column
#pragma unroll
    for (int k0 = 0; k0 < K; k0 += 4) {
      const int kk = k0 + 2 * hf;
      v2f a, b;
      a.x = aRow[kk] * rdeg;
      a.y = aRow[kk + 1] * rdeg;
      b.x = sWl[kk * 64 + cb];
      b.y = sWl[(kk + 1) * 64 + cb];
      c = __builtin_amdgcn_wmma_f32_16x16x4_f32(false, a, false, b, (short)0, c, false, false);
    }
#pragma unroll
    for (int k0 = 0; k0 < K; k0 += 4) {
      const int kk = k0 + 2 * hf;
      v2f a, b;
      a.x = hRow[kk];
      a.y = hRow[kk + 1];
      b.x = sWr[kk * 64 + cb];
      b.y = sWr[(kk + 1) * 64 + cb];
      c = __builtin_amdgcn_wmma_f32_16x16x4_f32(false, a, false, b, (short)0, c, false, false);
    }
    const float bb = bias[cb];
#pragma unroll
    for (int v = 0; v < 8; ++v)
      sOut[wave][(v + 8 * hf) * STR + cb] = c[v] + bb;
  }
  __syncthreads();

  // LayerNorm statistics: one lane per row (rows 0..15).
  if (lane < 16) {
    float s = 0.f, s2 = 0.f;
    for (int cc = 0; cc < 64; ++cc) {
      float vv = sOut[wave][lane * STR + cc];
      s += vv;
      s2 += vv * vv;
    }
    float mu = s * (1.0f / 64.0f);
    float var = s2 * (1.0f / 64.0f) - mu * mu;
    sStat[wave][lane] = mu;
    sStat[wave][16 + lane] = rsqrtf(var + 1e-5f);
  }
  __syncthreads();

  if (active) {
    for (int m = 0; m < 16; ++m) {
      const float mu = sStat[wave][m];
      const float rs = sStat[wave][16 + m];
      for (int cc = lane; cc < 64; cc += 32) {
        float vv = (sOut[wave][m * STR + cc] - mu) * rs * gamma[cc] + beta[cc];
        out[(node0 + m) * 64 + cc] = fmaxf(vv, 0.0f);
      }
    }
  }
}

// Layer 3: logits[N,2] = (agg/deg)@Wl3 + b3 + h@Wr3 (no LN/ReLU).
__global__ void sage_out(const float* __restrict__ agg, const float* __restrict__ deg,
                         const float* __restrict__ h, const float* __restrict__ Wl,
                         const float* __restrict__ Wr, const float* __restrict__ b,
                         float* __restrict__ out, int n) {
  long i = (long)blockIdx.x * blockDim.x + threadIdx.x;
  if (i >= n) return;
  float rdeg = 1.0f / fmaxf(deg[i], 1.0f);
  float o0 = b[0], o1 = b[1];
  const float* aRow = agg + i * 64;
  const float* hRow = h + i * 64;
#pragma unroll 8
  for (int k = 0; k < 64; ++k) {
    float a = aRow[k] * rdeg;
    float hh = hRow[k];
    o0 += a * Wl[k * 2 + 0] + hh * Wr[k * 2 + 0];
    o1 += a * Wl[k * 2 + 1] + hh * Wr[k * 2 + 1];
  }
  out[i * 2 + 0] = o0;
  out[i * 2 + 1] = o1;
}

static inline unsigned nblk(long n, int b) { return (unsigned)((n + b - 1) / b); }

extern "C" void kernel_launch(void* const* d_in, const int* in_sizes, int n_in,
                              void* d_out, int out_size, void* d_ws, size_t ws_size,
                              hipStream_t stream) {
  const float* x = (const float*)d_in[0];
  const long long* ei = (const long long*)d_in[1];
  const float* Wl1 = (const float*)d_in[2];
  const float* Wr1 = (const float*)d_in[3];
  const float* b1  = (const float*)d_in[4];
  const float* g1  = (const float*)d_in[5];
  const float* be1 = (const float*)d_in[6];
  const float* Wl2 = (const float*)d_in[7];
  const float* Wr2 = (const float*)d_in[8];
  const float* b2  = (const float*)d_in[9];
  const float* g2  = (const float*)d_in[10];
  const float* be2 = (const float*)d_in[11];
  const float* Wl3 = (const float*)d_in[12];
  const float* Wr3 = (const float*)d_in[13];
  const float* b3  = (const float*)d_in[14];

  const int N = in_sizes[0] / 32;   // IN_CH = 32
  const int E = in_sizes[1] / 2;    // edge_index is [2, E] int64
  const long long* src = ei;
  const long long* tgt = ei + E;

  // workspace: deg[N] (padded), agg[N*64], h1[N*64], h2[N*64]  (~77 MB)
  float* deg  = (float*)d_ws;
  float* aggA = deg + (((long)N + 255) & ~255L);
  float* h1   = aggA + (long)N * 64;
  float* h2   = h1 + (long)N * 64;

  const unsigned cblk = nblk(((N + 15) / 16 + 3) / 4, 1);  // combine blocks
  const unsigned combineBlocks = (unsigned)(((N + 15) / 16 + 3) / 4);
  (void)cblk; (void)ws_size; (void)n_in; (void)out_size;

  // degree (shared by all layers)
  zero_f32<<<nblk(N, 256), 256, 0, stream>>>(deg, N);
  deg_count<<<nblk(E, 256), 256, 0, stream>>>(tgt, deg, E);

  // ---- layer 1: IN_CH=32 -> HID=64 ----
  zero_f32<<<nblk((long)N * 32, 256), 256, 0, stream>>>(aggA, (long)N * 32);
  scatter_add<32><<<nblk((long)E * 8, 256), 256, 0, stream>>>(x, src, tgt, aggA, E);
  sage_combine<32><<<combineBlocks, 128, 0, stream>>>(aggA, deg, x, Wl1, Wr1, b1, g1, be1, h1, N);

  // ---- layer 2: 64 -> 64 ----
  zero_f32<<<nblk((long)N * 64, 256), 256, 0, stream>>>(aggA, (long)N * 64);
  scatter_add<64><<<nblk((long)E * 16, 256), 256, 0, stream>>>(h1, src, tgt, aggA, E);
  sage_combine<64><<<combineBlocks, 128, 0, stream>>>(aggA, deg, h1, Wl2, Wr2, b2, g2, be2, h2, N);

  // ---- layer 3: 64 -> 2 ----
  zero_f32<<<nblk((long)N * 64, 256), 256, 0, stream>>>(aggA, (long)N * 64);
  scatter_add<64><<<nblk((long)E * 16, 256), 256, 0, stream>>>(h2, src, tgt, aggA, E);
  sage_out<<<nblk(N, 256), 256, 0, stream>>>(aggA, deg, h2, Wl3, Wr3, b3, (float*)d_out, N);
}